// PGWAnchorFreeModule_32710470926886
// MI455X (gfx1250) — compile-verified
//
#include <hip/hip_runtime.h>
#include <hip/hip_bf16.h>
#include <math.h>

#define TOPK 9
#define EPSV 1e-10f
#define IOU_EPS 1e-6f

typedef float v2f __attribute__((ext_vector_type(2)));
typedef float v8f __attribute__((ext_vector_type(8)));

// ---------------- Phase 0: zero the output (harness poisons d_out) ----------
__global__ void zero_kernel(float* __restrict__ w, int n) {
  int i = blockIdx.x * blockDim.x + threadIdx.x;
  if (i < n) w[i] = 0.0f;
}

// ---------------- Phase 1: per-gt top-9 over all points ---------------------
// One block (256 threads = 8 wave32) per gt. Per-thread register top-9 with
// fully unrolled insertion (constant indices -> no scratch), then LDS merge
// tree of sorted lists. Ties prefer lower point index (matches stable top_k).
__global__ __launch_bounds__(256) void topk_kernel(
    const float* __restrict__ bbox_preds, const float* __restrict__ cls_scores,
    const float* __restrict__ gt_bboxes,  const int* __restrict__ gt_labels,
    int* __restrict__ topk_out, int N, int C)
{
  __shared__ float svals[256 * TOPK];
  __shared__ int   sidx [256 * TOPK];
  const int g   = blockIdx.x;
  const int tid = threadIdx.x;
  const float gx1 = gt_bboxes[g * 4 + 0], gy1 = gt_bboxes[g * 4 + 1];
  const float gx2 = gt_bboxes[g * 4 + 2], gy2 = gt_bboxes[g * 4 + 3];
  const float areag = (gx2 - gx1) * (gy2 - gy1);
  const int label = gt_labels[g];

  float tv[TOPK]; int ti[TOPK];
#pragma unroll
  for (int i = 0; i < TOPK; ++i) { tv[i] = -1.0f; ti[i] = 0x7fffffff; }

  for (int p = tid; p < N; p += 256) {
    float4 bp = reinterpret_cast<const float4*>(bbox_preds)[p];
    float ltx = fmaxf(bp.x, gx1), lty = fmaxf(bp.y, gy1);
    float rbx = fminf(bp.z, gx2), rby = fminf(bp.w, gy2);
    float wx = fmaxf(rbx - ltx, 0.0f), wy = fmaxf(rby - lty, 0.0f);
    float inter = wx * wy;
    float areap = (bp.z - bp.x) * (bp.w - bp.y);
    float uni   = areap + areag - inter;
    float iou   = inter / fmaxf(uni, IOU_EPS);
    float x   = cls_scores[(size_t)p * C + label];
    float cls = 1.0f / (1.0f + expf(-x));
    float score = 0.0f;
    if (iou > 0.0f)
      score = exp2f(0.8f * log2f(iou) + 0.2f * log2f(cls));
    if (score > tv[TOPK - 1]) {        // sift-in, keeps list sorted descending
      float s = score; int si = p;
#pragma unroll
      for (int j = 0; j < TOPK; ++j) {
        if (s > tv[j]) {
          float ft = tv[j]; tv[j] = s;  s  = ft;
          int   it = ti[j]; ti[j] = si; si = it;
        }
      }
    }
  }
#pragma unroll
  for (int j = 0; j < TOPK; ++j) { svals[tid * TOPK + j] = tv[j]; sidx[tid * TOPK + j] = ti[j]; }
  __syncthreads();

  for (int off = 128; off >= 1; off >>= 1) {
    if (tid < off) {
      const int pa = tid * TOPK, pb = (tid + off) * TOPK;
      float ov[TOPK]; int oi[TOPK];
      int ia = 0, ib = 0;                  // ia+ib==k<=8 -> never out of range
#pragma unroll
      for (int k = 0; k < TOPK; ++k) {
        float va = svals[pa + ia], vb = svals[pb + ib];
        int   xa = sidx [pa + ia], xb = sidx [pb + ib];
        bool pick = (va > vb) || ((va == vb) && (xa < xb));
        ov[k] = pick ? va : vb; oi[k] = pick ? xa : xb;
        ia += pick ? 1 : 0;     ib += pick ? 0 : 1;
      }
#pragma unroll
      for (int k = 0; k < TOPK; ++k) { svals[pa + k] = ov[k]; sidx[pa + k] = oi[k]; }
    }
    __syncthreads();
  }
  if (tid < TOPK) topk_out[g * TOPK + tid] = sidx[tid];
}

// ---------------- Phase 2: covariance via WMMA + Mahalanobis + scatter ------
// One wave32 per 8 gts. sigma_g = diff_g^T diff_g batched as a 16x16 Gram
// matrix via V_WMMA_F32_16X16X4_F32 (rows M=2g+i), K-chunked over the 9
// candidates. The f32 A(16x4) and B(4x16) lane layouts mirror (K=r+2*half,
// M/N=lane&15), so the same registers serve as A and B for the Gram product.
__global__ __launch_bounds__(32) void pgw_kernel(
    const float* __restrict__ points, const float* __restrict__ gt_bboxes,
    const int* __restrict__ topk_in, float* __restrict__ w)
{
  __shared__ float cpx[8 * TOPK], cpy[8 * TOPK];
  __shared__ int   cidx[8 * TOPK];
  __shared__ float mux[8], muy[8];
  __shared__ float sig[8][4];
  __shared__ float inv[8][4];

  const int t  = threadIdx.x;
  const int g0 = blockIdx.x * 8;

  for (int j = t; j < 8 * TOPK; j += 32) {
    int idx = topk_in[g0 * TOPK + j];
    cidx[j] = idx;
    cpx[j]  = points[idx * 2 + 0];
    cpy[j]  = points[idx * 2 + 1];
  }
  __syncthreads();

  if (t < 8) {
    float sx = 0.0f, sy = 0.0f;
#pragma unroll
    for (int k = 0; k < TOPK; ++k) { sx += cpx[t * TOPK + k]; sy += cpy[t * TOPK + k]; }
    mux[t] = sx * (1.0f / TOPK); muy[t] = sy * (1.0f / TOPK);
  }
  __syncthreads();

  // Build A fragment: lane holds A[M][K], M=lane&15, K=r+2*(lane>>4), r=0,1.
  const int M = t & 15, half = t >> 4;
  const int ga = M >> 1, dim = M & 1;
  const float mu = (dim == 0) ? mux[ga] : muy[ga];
  v8f acc = {0.f, 0.f, 0.f, 0.f, 0.f, 0.f, 0.f, 0.f};
#pragma unroll
  for (int c = 0; c < 3; ++c) {           // K-chunks cover candidates 0..11
    int k0 = c * 4 + 2 * half;
    float e0 = 0.0f, e1 = 0.0f;
    if (k0 < TOPK)
      e0 = ((dim == 0) ? cpx[ga * TOPK + k0] : cpy[ga * TOPK + k0]) - mu;
    if (k0 + 1 < TOPK)
      e1 = ((dim == 0) ? cpx[ga * TOPK + k0 + 1] : cpy[ga * TOPK + k0 + 1]) - mu;
    v2f a; a.x = e0; a.y = e1;
    acc = __builtin_amdgcn_wmma_f32_16x16x4_f32(
        /*neg_a=*/false, a, /*neg_b=*/false, a,
        /*c_mod=*/(short)0, acc, /*reuse_a=*/false, /*reuse_b=*/false);
  }

  // Extract 2x2 diagonal blocks. D layout: lane l holds column N=l&15,
  // rows M = r + 8*(l>>4). Lane is useful iff its half holds rows 2g,2g+1.
  {
    const int Nc = t & 15, gcol = Nc >> 1, jj = Nc & 1;
    if (half == (Nc >> 3)) {
      const int r0 = (2 * gcol) & 7;
      float e_lo = (r0 == 0) ? acc[0] : (r0 == 2) ? acc[2] : (r0 == 4) ? acc[4] : acc[6];
      float e_hi = (r0 == 0) ? acc[1] : (r0 == 2) ? acc[3] : (r0 == 4) ? acc[5] : acc[7];
      sig[gcol][0 * 2 + jj] = e_lo * (1.0f / TOPK);
      sig[gcol][1 * 2 + jj] = e_hi * (1.0f / TOPK);
    }
  }
  __syncthreads();

  if (t < 8) {
    float a_ = sig[t][0], b_ = sig[t][1], c_ = sig[t][2], d_ = sig[t][3];
    float det = a_ * d_ - b_ * c_;
    float r   = 1.0f / (det + EPSV);
    inv[t][0] =  d_ * r; inv[t][1] = -b_ * r;
    inv[t][2] = -c_ * r; inv[t][3] =  a_ * r;
  }
  __syncthreads();

  for (int j = t; j < 8 * TOPK; j += 32) {
    const int g = j / TOPK, gg = g0 + g;
    float dx = cpx[j] - mux[g], dy = cpy[j] - muy[g];
    float maha = dx * (inv[g][0] * dx + inv[g][1] * dy) +
                 dy * (inv[g][2] * dx + inv[g][3] * dy);
    float wgt = expf(-0.5f * maha);
    float gx1 = gt_bboxes[gg * 4 + 0], gy1 = gt_bboxes[gg * 4 + 1];
    float gx2 = gt_bboxes[gg * 4 + 2], gy2 = gt_bboxes[gg * 4 + 3];
    // reference: cy = cp[:,:,0], cx = cp[:,:,1] (note the swap)
    float cy = cpx[j], cx = cpy[j];
    bool valid = (cx - gx1 > EPSV) && (cy - gy1 > EPSV) &&
                 (gx2 - cx > EPSV) && (gy2 - cy > EPSV);
    float wv = valid ? wgt : 0.0f;
    // wv >= 0 and w pre-zeroed: unsigned-bit max == float max
    atomicMax(reinterpret_cast<unsigned int*>(&w[cidx[j]]), __float_as_uint(wv));
  }
}

// ---------------------------------------------------------------------------
extern "C" void kernel_launch(void* const* d_in, const int* in_sizes, int n_in,
                              void* d_out, int out_size, void* d_ws, size_t ws_size,
                              hipStream_t stream) {
  const float* points     = (const float*)d_in[0];
  const float* cls_scores = (const float*)d_in[1];
  const float* bbox_preds = (const float*)d_in[2];
  const float* gt_bboxes  = (const float*)d_in[3];
  const int*   gt_labels  = (const int*)d_in[4];

  const int N = in_sizes[0] / 2;    // 65536 points (x,y)
  const int G = in_sizes[4];        // 256 gts
  const int C = in_sizes[1] / N;    // 80 classes

  float* w    = (float*)d_out;
  int*   topk = (int*)d_ws;         // G*TOPK ints of scratch

  zero_kernel<<<(N + 255) / 256, 256, 0, stream>>>(w, N);
  topk_kernel<<<G, 256, 0, stream>>>(bbox_preds, cls_scores, gt_bboxes,
                                     gt_labels, topk, N, C);
  pgw_kernel<<<G / 8, 32, 0, stream>>>(points, gt_bboxes, topk, w);
}